// CRF_10015863734521
// MI455X (gfx1250) — compile-verified
//
#include <hip/hip_runtime.h>

// CRF forward scan for MI455X (gfx1250, wave32).
// Exp-domain reformulation turns the per-step logsumexp j-reduction into a
// GEMM against the constant matrix P = exp(transition), executed with
// v_wmma_f32_16x16x4_f32. P's B-fragments are hoisted into registers before
// the 512-step scan. State lives in LDS; next step's features are prefetched.

typedef __attribute__((ext_vector_type(2))) float v2f;
typedef __attribute__((ext_vector_type(8))) float v8f;

#define TT     48      // tag count
#define SS     512     // sequence length
#define BATCH  128
#define BTILE  16      // batch rows per workgroup
#define LDSP   52      // padded LDS row stride (even -> 8B-aligned float2 loads)
#define MINV   1e-4f
#define STARTI 46
#define ENDI   47

__global__ __launch_bounds__(96) void crf_fwd_kernel(
    const float* __restrict__ features,    // [B, S, T]
    const float* __restrict__ transition,  // [T, T]
    float* __restrict__ out)               // [B]
{
    __shared__ float trans_s[TT][LDSP];    // transition (log domain)
    __shared__ float fv   [BTILE][LDSP];   // forward variable
    __shared__ float featb[BTILE][LDSP];   // current step's feature tile
    __shared__ float ms   [BTILE][LDSP];   // m[b,i] = max_j(fv[b,j]+trans[i,j])
    __shared__ float us   [BTILE][LDSP];   // u[b,j] = exp(fv - feat - m)

    const int tid  = threadIdx.x;
    const int lane = tid & 31;
    const int wave = tid >> 5;            // 0..2 -> i-tile
    const int half = lane >> 4;           // 0: K={0,1}, 1: K={2,3} per 4-group
    const int b0   = blockIdx.x * BTILE;

    // ---- load transition into LDS (2304 elements, 24 per thread) ----
    #pragma unroll
    for (int p = 0; p < 24; ++p) {
        int idx = tid + p * 96;
        trans_s[idx / TT][idx % TT] = transition[idx];
    }
    // ---- init fv0: MIN everywhere except 0 at START column ----
    #pragma unroll
    for (int p = 0; p < 8; ++p) {
        int idx = tid + p * 96;
        fv[idx / TT][idx % TT] = ((idx % TT) == STARTI) ? 0.0f : MINV;
    }
    __syncthreads();

    // ---- hoist constant WMMA B-fragments: B[k=j][n=i] = exp(trans[i][j]) ----
    // 32-bit B 4x16 layout: lanes 0-15 hold K=0,1 (vgpr0,1) for N=lane;
    // lanes 16-31 hold K=2,3 for N=lane-16.
    const int nI = wave * 16 + (lane & 15);   // this lane's output column i
    v2f bfrag[12];
    #pragma unroll
    for (int k = 0; k < 12; ++k) {
        int kk = k * 4 + half * 2;
        bfrag[k].x = __expf(trans_s[nI][kk]);
        bfrag[k].y = __expf(trans_s[nI][kk + 1]);
    }

    const size_t fbase = (size_t)b0 * SS * TT;
    const int mb = tid / 6;          // batch row for max-plus phase
    const int mg = tid % 6;          // i-group (8 i's each)

    #pragma unroll 1
    for (int t = 0; t < SS; ++t) {
        // ---- stage feature tile for step t into LDS ----
        const float* fptr = features + fbase + (size_t)t * TT;
        #pragma unroll
        for (int p = 0; p < 8; ++p) {
            int idx = tid + p * 96;
            int b = idx / TT, j = idx % TT;
            featb[b][j] = fptr[(size_t)b * SS * TT + j];
        }
        // prefetch next step's tile (16 rows x 192B, 6 chunks/row)
        if (t + 1 < SS) {
            const float* nf = fptr + TT;
            __builtin_prefetch(nf + (size_t)(tid & 15) * SS * TT + (tid >> 4) * 8, 0, 3);
        }

        // ---- m[b,i] = max_j(fv[b,j] + trans[i,j]) : max-plus on VALU ----
        float best[8];
        #pragma unroll
        for (int q = 0; q < 8; ++q) best[q] = -3.4e38f;
        #pragma unroll 4
        for (int j = 0; j < TT; ++j) {
            float f = fv[mb][j];
            #pragma unroll
            for (int q = 0; q < 8; ++q)
                best[q] = fmaxf(best[q], f + trans_s[mg * 8 + q][j]);
        }
        #pragma unroll
        for (int q = 0; q < 8; ++q) ms[mb][mg * 8 + q] = best[q];
        __syncthreads();   // feat tile + m visible

        // ---- u[b,j] = exp(fv - feat - m) ----
        #pragma unroll
        for (int p = 0; p < 8; ++p) {
            int idx = tid + p * 96;
            int b = idx / TT, j = idx % TT;
            us[b][j] = __expf(fv[b][j] - featb[b][j] - ms[b][j]);
        }
        __syncthreads();   // u ready; all fv reads for this step done

        // ---- S[b,i] = sum_j u[b,j] * P[i,j] : 12x chained f32 WMMA (K=48) ----
        // 32-bit A 16x4 layout: lane<16 holds M=lane, K=4k+{0,1};
        // lane>=16 holds M=lane-16, K=4k+{2,3}.
        v8f acc = {};
        const int M = lane & 15;
        #pragma unroll
        for (int k = 0; k < 12; ++k) {
            int kk = k * 4 + half * 2;
            v2f a = *(const v2f*)&us[M][kk];     // 8B-aligned (kk even, LDSP even)
            acc = __builtin_amdgcn_wmma_f32_16x16x4_f32(
                false, a, false, bfrag[k], (short)0, acc, false, false);
        }

        // ---- new_fv[b,i] = m[b,i] + 2*feat[b,i] + log(S[b,i]) ----
        // D layout: vgpr r, lane -> (M = r + 8*half, N = lane&15)
        #pragma unroll
        for (int r = 0; r < 8; ++r) {
            int b = r + half * 8;
            fv[b][nI] = ms[b][nI] + 2.0f * featb[b][nI] + __logf(acc[r]);
        }
        __syncthreads();   // fv updated before next step reads it
    }

    // ---- terminal: sc[b,j] = fv[b,j] + trans[END][j]; logsumexp over j ----
    if (tid < BTILE) {
        int b = tid;
        float mx = -3.4e38f;
        #pragma unroll 4
        for (int j = 0; j < TT; ++j)
            mx = fmaxf(mx, fv[b][j] + trans_s[ENDI][j]);
        float sum = 0.0f;
        #pragma unroll 4
        for (int j = 0; j < TT; ++j)
            sum += __expf(fv[b][j] + trans_s[ENDI][j] - mx);
        out[b0 + b] = mx + __logf(sum);
    }
}

extern "C" void kernel_launch(void* const* d_in, const int* in_sizes, int n_in,
                              void* d_out, int out_size, void* d_ws, size_t ws_size,
                              hipStream_t stream) {
    const float* features   = (const float*)d_in[0];   // [128, 512, 48] f32
    const float* transition = (const float*)d_in[1];   // [48, 48] f32
    float* out = (float*)d_out;                        // [128] f32
    crf_fwd_kernel<<<dim3(BATCH / BTILE), dim3(96), 0, stream>>>(features, transition, out);
}